// DBHDSNetLoss_39994735460482
// MI455X (gfx1250) — compile-verified
//
#include <hip/hip_runtime.h>
#include <math.h>

// ---------------- constants ----------------
#define NCLS 38
#define GAMMA 1.5f
#define ALPHA_F 0.25f
#define EPSF 1e-7f
#define B_ 16
#define N_GT 50

// accumulator layout in d_ws (floats)
#define ACC_OBJ   0   // [3]
#define ACC_BOX   3   // [3]
#define ACC_CLS   6   // [3]
#define ACC_NPOS  9   // [3]
#define ACC_BCE   12  // [1]
#define ACC_INTER 13  // [16]
#define ACC_SP    29  // [16]
#define ACC_ST    45  // [16]
#define ACC_CE    61  // [16]
#define ACC_EC    77  // [16]
#define ACC_TOTAL 128

typedef __attribute__((ext_vector_type(2))) float v2f;
typedef __attribute__((ext_vector_type(8))) float v8f;

__device__ __forceinline__ void atomAddF(float* p, float v) {
    __hip_atomic_fetch_add(p, v, __ATOMIC_RELAXED, __HIP_MEMORY_SCOPE_AGENT);
}
__device__ __forceinline__ float sigmoidf(float x) { return 1.0f / (1.0f + expf(-x)); }
__device__ __forceinline__ float softplusf(float x) { return fmaxf(x, 0.0f) + log1pf(expf(-fabsf(x))); }

// block reduction: returns full sum in thread 0 (blockDim.x multiple of 32, <=256)
__device__ __forceinline__ float blockReduce(float v, float* sbuf) {
    for (int o = 16; o > 0; o >>= 1) v += __shfl_down(v, o, 32);
    const int wave = threadIdx.x >> 5, lane = threadIdx.x & 31;
    if (lane == 0) sbuf[wave] = v;
    __syncthreads();
    float r = 0.0f;
    if (wave == 0) {
        r = (lane < (int)(blockDim.x >> 5)) ? sbuf[lane] : 0.0f;
        for (int o = 4; o > 0; o >>= 1) r += __shfl_down(r, o, 32);
    }
    __syncthreads();
    return r;
}

// ---------------- zero accumulators ----------------
__global__ void zero_ws_kernel(float* acc) {
    if (threadIdx.x < ACC_TOTAL) acc[threadIdx.x] = 0.0f;
}

// ---------------- obj full-map softplus reduction ----------------
__global__ void obj_reduce_kernel(const float* __restrict__ x, int n, float* acc, int accIdx) {
    __shared__ float sbuf[8];
    float s = 0.0f;
    for (int i = blockIdx.x * blockDim.x + threadIdx.x; i < n; i += gridDim.x * blockDim.x)
        s += softplusf(x[i]);
    float r = blockReduce(s, sbuf);
    if (threadIdx.x == 0) atomAddF(&acc[accIdx], r);
}

// ---------------- positive-cell losses (box CIoU, focal cls, obj correction) ----------------
__device__ __forceinline__ float ciou_loss(const float* d, float tx, float ty, float tw, float th_) {
    float px = d[0], py = d[1], pw = d[2], ph = d[3];
    float px1 = px - pw * 0.5f, py1 = py - ph * 0.5f, px2 = px + pw * 0.5f, py2 = py + ph * 0.5f;
    float tx1 = tx - tw * 0.5f, ty1 = ty - th_ * 0.5f, tx2 = tx + tw * 0.5f, ty2 = ty + th_ * 0.5f;
    float iw = fmaxf(fminf(px2, tx2) - fmaxf(px1, tx1), 0.0f);
    float ih = fmaxf(fminf(py2, ty2) - fmaxf(py1, ty1), 0.0f);
    float inter = iw * ih;
    float uni = pw * ph + tw * th_ - inter + EPSF;
    float iou = inter / uni;
    float cw = fmaxf(px2, tx2) - fminf(px1, tx1);
    float ch = fmaxf(py2, ty2) - fminf(py1, ty1);
    float c2 = cw * cw + ch * ch + EPSF;
    float rho2 = (px - tx) * (px - tx) + (py - ty) * (py - ty);
    float da = atanf(tw / (th_ + EPSF)) - atanf(pw / (ph + EPSF));
    float v = (4.0f / (float)(M_PI * M_PI)) * da * da;
    float alpha = v / (1.0f - iou + v + EPSF);
    return 1.0f - (iou - rho2 / c2 - alpha * v);
}

__global__ void poscell_kernel(const float* __restrict__ reg3, const float* __restrict__ obj3,
                               const float* __restrict__ cls3, const float* __restrict__ reg4,
                               const float* __restrict__ obj4, const float* __restrict__ cls4,
                               const float* __restrict__ reg5, const float* __restrict__ obj5,
                               const float* __restrict__ cls5, const float* __restrict__ gtb,
                               const int* __restrict__ gtc, float* acc) {
    int tid = blockIdx.x * blockDim.x + threadIdx.x;
    if (tid >= 48) return;
    const int s = tid / 16, b = tid % 16;
    const int Wd = (s == 0) ? 80 : (s == 1) ? 40 : 20;
    const int HW = Wd * Wd;
    const float lo = (s == 0) ? 0.00f : (s == 1) ? 0.10f : 0.25f;
    const float hi = (s == 0) ? 0.15f : (s == 1) ? 0.35f : 1.00f;
    const float* regp = (s == 0) ? reg3 : (s == 1) ? reg4 : reg5;
    const float* objp = (s == 0) ? obj3 : (s == 1) ? obj4 : obj5;
    const float* clsp = (s == 0) ? cls3 : (s == 1) ? cls4 : cls5;

    int cells[N_GT];
    for (int n = 0; n < N_GT; ++n) {
        const float* g = gtb + ((size_t)b * N_GT + n) * 4;
        float cx = g[0], cy = g[1], w = g[2], h = g[3];
        float ms = fmaxf(w, h);
        if (ms >= lo && ms < hi) {
            int gx = min(max((int)(cx * (float)Wd), 0), Wd - 1);
            int gy = min(max((int)(cy * (float)Wd), 0), Wd - 1);
            cells[n] = gy * Wd + gx;
        } else {
            cells[n] = -1;
        }
    }

    for (int n = 0; n < N_GT; ++n) {
        int cell = cells[n];
        if (cell < 0) continue;
        bool winner = true;                 // last valid GT per cell wins (scatter .set order)
        for (int n2 = n + 1; n2 < N_GT; ++n2)
            if (cells[n2] == cell) { winner = false; break; }
        if (!winner) continue;

        unsigned long long msk = 0ull;      // union of one-hot class targets at this cell
        for (int n2 = 0; n2 < N_GT; ++n2)
            if (cells[n2] == cell) msk |= (1ull << gtc[b * N_GT + n2]);

        // obj BCE correction at positive cell: bce(x,1) = softplus(x) - x
        atomAddF(&acc[ACC_OBJ + s], -objp[(size_t)b * HW + cell]);

        // box CIoU (decoded pred = sigmoid of reg logits, channel-major gather)
        float dec[4];
        for (int c = 0; c < 4; ++c)
            dec[c] = sigmoidf(regp[((size_t)b * 4 + c) * HW + cell]);
        const float* g = gtb + ((size_t)b * N_GT + n) * 4;
        atomAddF(&acc[ACC_BOX + s], ciou_loss(dec, g[0], g[1], g[2], g[3]));

        // focal over 38 classes at this cell
        float fs = 0.0f;
        for (int j = 0; j < NCLS; ++j) {
            float L = clsp[((size_t)b * NCLS + j) * HW + cell];
            float t = ((msk >> j) & 1ull) ? 1.0f : 0.0f;
            float p = sigmoidf(L);
            float ce = fmaxf(L, 0.0f) - L * t + log1pf(expf(-fabsf(L)));
            float pt = p * t + (1.0f - p) * (1.0f - t);
            float w = powf(1.0f - pt, GAMMA);
            float at = ALPHA_F * t + (1.0f - ALPHA_F) * (1.0f - t);
            fs += at * w * ce;
        }
        atomAddF(&acc[ACC_CLS + s], fs);
        atomAddF(&acc[ACC_NPOS + s], 1.0f);
    }
}

// ---------------- seg loss: streaming 105 MB reduction (bandwidth-bound) ----------------
#define SEG_BPB 200          // blocks per batch
#define SEG_F4_PER_BATCH 204800
__global__ void seg_kernel(const float* __restrict__ proto, const float* __restrict__ gtm, float* acc) {
    __shared__ float sbuf[8];
    const int batch = blockIdx.x / SEG_BPB;
    const int part = blockIdx.x % SEG_BPB;
    const float4* P = (const float4*)proto;
    const float4* T = (const float4*)gtm;
    const long base = (long)batch * SEG_F4_PER_BATCH + (long)part * 1024;
    float bce = 0.0f, in = 0.0f, sp = 0.0f, st = 0.0f;
#pragma unroll
    for (int k = 0; k < 4; ++k) {
        long i = base + threadIdx.x + k * 256;
        if (k < 3) { __builtin_prefetch(&P[i + 256], 0, 0); __builtin_prefetch(&T[i + 256], 0, 0); }
        float4 pv = P[i];
        float4 tv = T[i];
        float xs[4] = {pv.x, pv.y, pv.z, pv.w};
        float ts[4] = {tv.x, tv.y, tv.z, tv.w};
#pragma unroll
        for (int c = 0; c < 4; ++c) {
            float p = sigmoidf(xs[c]);
            p = fminf(fmaxf(p, EPSF), 1.0f - EPSF);
            float t = ts[c];
            bce += -(t * logf(p) + (1.0f - t) * logf(1.0f - p));
            in += p * t;
            sp += p;
            st += t;
        }
    }
    float r;
    r = blockReduce(bce, sbuf); if (threadIdx.x == 0) atomAddF(&acc[ACC_BCE], r);
    r = blockReduce(in, sbuf);  if (threadIdx.x == 0) atomAddF(&acc[ACC_INTER + batch], r);
    r = blockReduce(sp, sbuf);  if (threadIdx.x == 0) atomAddF(&acc[ACC_SP + batch], r);
    r = blockReduce(st, sbuf);  if (threadIdx.x == 0) atomAddF(&acc[ACC_ST + batch], r);
}

// ---------------- hazard losses: probs(16x4) @ PENALTY^T via V_WMMA_F32_16X16X4_F32 ----------------
__global__ void hazard_kernel(const float* __restrict__ hz, const int* __restrict__ tiers, float* acc) {
    const int lane = threadIdx.x; // one wave32
    __shared__ float probsS[16][4];
    __shared__ int tierS[16];
    if (lane < 16) {
        float x0 = hz[lane * 4 + 0], x1 = hz[lane * 4 + 1];
        float x2 = hz[lane * 4 + 2], x3 = hz[lane * 4 + 3];
        float m = fmaxf(fmaxf(x0, x1), fmaxf(x2, x3));
        float e0 = expf(x0 - m), e1 = expf(x1 - m), e2 = expf(x2 - m), e3 = expf(x3 - m);
        float lse = m + logf(e0 + e1 + e2 + e3);
        probsS[lane][0] = expf(x0 - lse);
        probsS[lane][1] = expf(x1 - lse);
        probsS[lane][2] = expf(x2 - lse);
        probsS[lane][3] = expf(x3 - lse);
        int t = tiers[lane];
        tierS[lane] = t;
        float xt = (t == 0) ? x0 : (t == 1) ? x1 : (t == 2) ? x2 : x3;
        acc[ACC_CE + lane] = lse - xt; // -log_softmax at target tier
    }
    __syncthreads();

    const float PEN[4][4] = {{0.f, 1.f, 2.f, 4.f}, {2.f, 0.f, 1.f, 2.f},
                             {4.f, 2.f, 0.f, 1.f}, {8.f, 4.f, 2.f, 0.f}};
    const int m16 = lane & 15;
    const int kb = (lane < 16) ? 0 : 2; // ISA 16x4 f32 A layout: lanes 0-15 K=0/1, lanes 16-31 K=2/3
    v2f a, bm;
    a.x = probsS[m16][kb];
    a.y = probsS[m16][kb + 1];
    bm.x = (m16 < 4) ? PEN[m16][kb] : 0.0f;     // B[k][n] = PENALTY[n][k], cols 4..15 zero
    bm.y = (m16 < 4) ? PEN[m16][kb + 1] : 0.0f;
    v8f c = {0.f, 0.f, 0.f, 0.f, 0.f, 0.f, 0.f, 0.f};
    c = __builtin_amdgcn_wmma_f32_16x16x4_f32(false, a, false, bm, (short)0, c, false, false);

    // C/D 16x16 f32 layout: vgpr v -> row M = v (+8 for lanes 16-31), N = lane%16
#pragma unroll
    for (int v = 0; v < 8; ++v) {
        int mrow = v + ((lane >= 16) ? 8 : 0);
        if (m16 == tierS[mrow]) acc[ACC_EC + mrow] = c[v]; // exp_cost[b] = D[b][tier[b]]
    }
}

// ---------------- finalize ----------------
__global__ void finalize_kernel(const float* __restrict__ acc, float* __restrict__ out) {
    if (threadIdx.x != 0 || blockIdx.x != 0) return;
    const float denom[3] = {16.0f * 6400.0f, 16.0f * 1600.0f, 16.0f * 400.0f};
    float tb = 0.0f, to = 0.0f, tc = 0.0f;
    for (int s = 0; s < 3; ++s) {
        float npos = fmaxf(acc[ACC_NPOS + s], 1.0f);
        to += acc[ACC_OBJ + s] / denom[s];
        tb += acc[ACC_BOX + s] / npos;
        tc += acc[ACC_CLS + s] / (npos * (float)NCLS);
    }
    tb /= 3.0f; to /= 3.0f; tc /= 3.0f;
    float bce = acc[ACC_BCE] / (16.0f * 819200.0f);
    float dsum = 0.0f;
    for (int b = 0; b < B_; ++b)
        dsum += 1.0f - (2.0f * acc[ACC_INTER + b] + 1.0f) / (acc[ACC_SP + b] + acc[ACC_ST + b] + 1.0f);
    float ts = (bce + dsum / 16.0f) * 0.5f;
    float th = 0.0f, thier = 0.0f;
    for (int b = 0; b < B_; ++b) {
        float ce = acc[ACC_CE + b];
        th += ce;
        thier += acc[ACC_EC + b] * ce;
    }
    th /= 16.0f; thier /= 16.0f;
    out[0] = 5.0f * tb + 1.0f * to + 1.0f * tc + 2.0f * ts + 1.0f * th + 0.5f * thier;
}

// ---------------- launcher ----------------
extern "C" void kernel_launch(void* const* d_in, const int* in_sizes, int n_in,
                              void* d_out, int out_size, void* d_ws, size_t ws_size,
                              hipStream_t stream) {
    const float* reg3 = (const float*)d_in[0];
    const float* obj3 = (const float*)d_in[1];
    const float* cls3 = (const float*)d_in[2];
    const float* reg4 = (const float*)d_in[3];
    const float* obj4 = (const float*)d_in[4];
    const float* cls4 = (const float*)d_in[5];
    const float* reg5 = (const float*)d_in[6];
    const float* obj5 = (const float*)d_in[7];
    const float* cls5 = (const float*)d_in[8];
    const float* hz   = (const float*)d_in[9];
    const float* prot = (const float*)d_in[10];
    const float* gtm  = (const float*)d_in[11];
    const float* gtb  = (const float*)d_in[12];
    const int* gtc    = (const int*)d_in[13];
    const int* gth    = (const int*)d_in[14];
    float* acc = (float*)d_ws;
    float* out = (float*)d_out;

    zero_ws_kernel<<<1, 128, 0, stream>>>(acc);

    obj_reduce_kernel<<<50, 256, 0, stream>>>(obj3, 16 * 6400, acc, ACC_OBJ + 0);
    obj_reduce_kernel<<<13, 256, 0, stream>>>(obj4, 16 * 1600, acc, ACC_OBJ + 1);
    obj_reduce_kernel<<<4, 256, 0, stream>>>(obj5, 16 * 400, acc, ACC_OBJ + 2);

    poscell_kernel<<<1, 64, 0, stream>>>(reg3, obj3, cls3, reg4, obj4, cls4,
                                         reg5, obj5, cls5, gtb, gtc, acc);

    seg_kernel<<<16 * SEG_BPB, 256, 0, stream>>>(prot, gtm, acc);

    hazard_kernel<<<1, 32, 0, stream>>>(hz, gth, acc);

    finalize_kernel<<<1, 32, 0, stream>>>(acc, out);
}